// SCOUT_MTP_23862838297011
// MI455X (gfx1250) — compile-verified
//
#include <hip/hip_runtime.h>
#include <hip/hip_bf16.h>

typedef __attribute__((ext_vector_type(2))) float v2f;
typedef __attribute__((ext_vector_type(8))) float v8f;

#define N_NODES 16384
#define N_EDGES 262144
#define DIM     256

// ---------------------------------------------------------------------------
// Kernel 1: dual GEMM  hs = h @ W_self (grid.y==0), z = h @ W_func (grid.y==1)
// V_WMMA_F32_16X16X4_F32. Block = 256 threads (8 wave32).
// Each block: 2 M-tiles (16 rows) x 4 N-strips (64 cols).
// W staged in LDS in 32-row K-slabs, PAIR-INTERLEAVED so each B fragment
// ({W[k][col], W[k+1][col]}) is a single ds_load_b64:
//   lds[(k>>1)*512 + col*2 + (k&1)] = W[slab+k][col]
// ---------------------------------------------------------------------------
__global__ void gat_gemm_dual(const float* __restrict__ h,
                              const float* __restrict__ Wself,
                              const float* __restrict__ Wfunc,
                              float* __restrict__ hs,
                              float* __restrict__ z)
{
    __shared__ float ldsW[32 * 256];   // one K-slab of W, pair-interleaved

    const float* W   = blockIdx.y ? Wfunc : Wself;
    float*       out = blockIdx.y ? z    : hs;

    const int wave = threadIdx.x >> 5;
    const int lane = threadIdx.x & 31;

    const int mtile  = blockIdx.x * 2 + (wave >> 2);   // 16-row tile
    const int nstrip = wave & 3;                        // 64-col strip
    const int mbase  = mtile * 16;
    const int nbase  = nstrip * 64;

    const int mrow  = mbase + (lane & 15);
    const int khalf = (lane >> 4) * 2;                  // 0 or 2 within K=4 chunk
    const int bcol  = nbase + (lane & 15);

    v8f acc0 = {}, acc1 = {}, acc2 = {}, acc3 = {};

    for (int slab = 0; slab < DIM; slab += 32) {
        __syncthreads();
        // cooperative pair-interleaved copy of W[slab..slab+31][0..255]
        const float4* gW = (const float4*)(W + slab * DIM);
        #pragma unroll
        for (int r = 0; r < 8; ++r) {
            const int idx4 = threadIdx.x + r * 256;     // 0..2047 (64 float4/row)
            const int k    = idx4 >> 6;                 // row within slab
            const int col  = (idx4 & 63) * 4;
            float4 v = gW[idx4];
            float* dst = &ldsW[(k >> 1) * 512 + col * 2 + (k & 1)];
            dst[0] = v.x; dst[2] = v.y; dst[4] = v.z; dst[6] = v.w;
        }
        __syncthreads();

        #pragma unroll
        for (int kc = 0; kc < 32; kc += 4) {
            const int k = slab + kc;
            // A fragment: lane holds h[mrow][k + khalf + {0,1}]
            v2f a = *(const v2f*)(h + (size_t)mrow * DIM + k + khalf);

            const int brow = kc + khalf;                // even row within slab
            const float* bbase = &ldsW[(brow >> 1) * 512];
            v2f b0 = *(const v2f*)(bbase + (bcol +  0) * 2);
            v2f b1 = *(const v2f*)(bbase + (bcol + 16) * 2);
            v2f b2 = *(const v2f*)(bbase + (bcol + 32) * 2);
            v2f b3 = *(const v2f*)(bbase + (bcol + 48) * 2);

            acc0 = __builtin_amdgcn_wmma_f32_16x16x4_f32(false, a, false, b0,
                                                         (short)0, acc0, false, false);
            acc1 = __builtin_amdgcn_wmma_f32_16x16x4_f32(false, a, false, b1,
                                                         (short)0, acc1, false, false);
            acc2 = __builtin_amdgcn_wmma_f32_16x16x4_f32(false, a, false, b2,
                                                         (short)0, acc2, false, false);
            acc3 = __builtin_amdgcn_wmma_f32_16x16x4_f32(false, a, false, b3,
                                                         (short)0, acc3, false, false);
        }
    }

    // C/D layout: VGPR r -> M = r + (lane>=16 ? 8 : 0), N = lane & 15
    const int ncol = lane & 15;
    const int mofs = (lane >> 4) * 8;
    #pragma unroll
    for (int r = 0; r < 8; ++r) {
        const size_t row = (size_t)(mbase + mofs + r) * DIM;
        out[row + nbase +  0 + ncol] = acc0[r];
        out[row + nbase + 16 + ncol] = acc1[r];
        out[row + nbase + 32 + ncol] = acc2[r];
        out[row + nbase + 48 + ncol] = acc3[r];
    }
}

// ---------------------------------------------------------------------------
// Kernel 2: per-node attention scalars zs = z . w_src, zd = z . w_dst
// One wave per node; shuffle reduction (wave32).
// ---------------------------------------------------------------------------
__global__ void gat_node_scalars(const float* __restrict__ z,
                                 const float* __restrict__ Watt,
                                 float* __restrict__ zs,
                                 float* __restrict__ zd)
{
    const int wave = threadIdx.x >> 5;
    const int lane = threadIdx.x & 31;
    const int node = blockIdx.x * 8 + wave;

    const float4* zr = (const float4*)(z + (size_t)node * DIM);
    const float4* ws = (const float4*)(Watt);
    const float4* wd = (const float4*)(Watt + DIM);

    float s = 0.f, d = 0.f;
    #pragma unroll
    for (int i = 0; i < 2; ++i) {
        float4 zv = zr[lane * 2 + i];
        float4 a  = ws[lane * 2 + i];
        float4 b  = wd[lane * 2 + i];
        s += zv.x * a.x + zv.y * a.y + zv.z * a.z + zv.w * a.w;
        d += zv.x * b.x + zv.y * b.y + zv.z * b.z + zv.w * b.w;
    }
    #pragma unroll
    for (int off = 16; off; off >>= 1) {
        s += __shfl_xor(s, off, 32);
        d += __shfl_xor(d, off, 32);
    }
    if (lane == 0) { zs[node] = s; zd[node] = d; }
}

// ---------------------------------------------------------------------------
// Kernel 3: edge logits + leaky_relu + segment max (monotonic-uint atomicMax)
// One wave per edge.
// ---------------------------------------------------------------------------
__device__ __forceinline__ unsigned f32_ordkey(float v) {
    unsigned u = __float_as_uint(v);
    return (v >= 0.f) ? (u | 0x80000000u) : ~u;
}

__global__ void gat_edge_logits(const float* __restrict__ edge_w,
                                const int* __restrict__ esrc,
                                const int* __restrict__ edst,
                                const float* __restrict__ Watt,
                                const float* __restrict__ zs,
                                const float* __restrict__ zd,
                                float* __restrict__ e,
                                unsigned* __restrict__ mkey,
                                int* __restrict__ deg)
{
    const int wave = threadIdx.x >> 5;
    const int lane = threadIdx.x & 31;
    const size_t eid = (size_t)blockIdx.x * 8 + wave;

    const float4* ew = (const float4*)(edge_w + eid * DIM);
    const float4* wv = (const float4*)(Watt + 2 * DIM);   // w_ew

    float s = 0.f;
    #pragma unroll
    for (int i = 0; i < 2; ++i) {
        float4 a = ew[lane * 2 + i];
        float4 b = wv[lane * 2 + i];
        s += a.x * b.x + a.y * b.y + a.z * b.z + a.w * b.w;
    }
    #pragma unroll
    for (int off = 16; off; off >>= 1) s += __shfl_xor(s, off, 32);

    if (lane == 0) {
        const int si = esrc[eid];
        const int di = edst[eid];
        float v = zs[si] + zd[di] + s;
        v = (v > 0.f) ? v : 0.01f * v;           // leaky_relu(0.01)
        e[eid] = v;
        atomicMax(mkey + di, f32_ordkey(v));
        atomicAdd(deg + di, 1);
    }
}

// ---------------------------------------------------------------------------
// Kernel 4: decode segment max (empty segment -> 0, non-finite guard)
// ---------------------------------------------------------------------------
__global__ void gat_decode_m(const unsigned* __restrict__ mkey,
                             const int* __restrict__ deg,
                             float* __restrict__ m)
{
    const int n = blockIdx.x * 256 + threadIdx.x;
    float v = 0.f;
    if (deg[n] > 0) {
        const unsigned k = mkey[n];
        v = (k & 0x80000000u) ? __uint_as_float(k & 0x7FFFFFFFu)
                              : __uint_as_float(~k);
        if (!isfinite(v)) v = 0.f;
    }
    m[n] = v;
}

// ---------------------------------------------------------------------------
// Kernel 5 (fused softmax-denominator + weighted scatter):
//   ex = exp(e - m[dst]);  den[dst] += ex;  agg[dst][:] += ex * z[src][:]
// One wave per edge; 8 f32 global atomics per lane (L2-resident).
// ---------------------------------------------------------------------------
__global__ void gat_edge_scatter(const float* __restrict__ e,
                                 const int* __restrict__ esrc,
                                 const int* __restrict__ edst,
                                 const float* __restrict__ m,
                                 const float* __restrict__ z,
                                 float* __restrict__ den,
                                 float* __restrict__ agg)
{
    const int wave = threadIdx.x >> 5;
    const int lane = threadIdx.x & 31;
    const size_t eid = (size_t)blockIdx.x * 8 + wave;

    const int si = esrc[eid];
    const int di = edst[eid];
    const float ex = expf(e[eid] - m[di]);

    if (lane == 0) atomicAdd(den + di, ex);

    const float4* zr = (const float4*)(z + (size_t)si * DIM);
    float*        ag = agg + (size_t)di * DIM;
    #pragma unroll
    for (int i = 0; i < 2; ++i) {
        float4 zv = zr[lane * 2 + i];
        const int c = (lane * 2 + i) * 4;
        atomicAdd(ag + c + 0, ex * zv.x);
        atomicAdd(ag + c + 1, ex * zv.y);
        atomicAdd(ag + c + 2, ex * zv.z);
        atomicAdd(ag + c + 3, ex * zv.w);
    }
}

// ---------------------------------------------------------------------------
// Kernel 6: out = h + relu( deg>0 ? hs + agg/den : h )
// One block per node, one thread per column.
// ---------------------------------------------------------------------------
__global__ void gat_finalize(const float* __restrict__ h,
                             const float* __restrict__ hs,
                             const float* __restrict__ agg,
                             const float* __restrict__ den,
                             const int* __restrict__ deg,
                             float* __restrict__ out)
{
    const size_t n   = blockIdx.x;
    const int    c   = threadIdx.x;
    const size_t idx = n * DIM + c;

    const float hv = h[idx];
    float v = (deg[n] > 0) ? (hs[idx] + agg[idx] / den[n]) : hv;
    v = (v > 0.f) ? v : 0.f;           // relu
    out[idx] = hv + v;
}

// ---------------------------------------------------------------------------
extern "C" void kernel_launch(void* const* d_in, const int* in_sizes, int n_in,
                              void* d_out, int out_size, void* d_ws, size_t ws_size,
                              hipStream_t stream)
{
    const float* h      = (const float*)d_in[0];
    const int*   esrc   = (const int*)  d_in[1];
    const int*   edst   = (const int*)  d_in[2];
    const float* edge_w = (const float*)d_in[3];
    const float* Wself  = (const float*)d_in[4];
    const float* Wfunc  = (const float*)d_in[5];
    const float* Watt   = (const float*)d_in[6];
    float*       out    = (float*)d_out;

    char* ws = (char*)d_ws;
    size_t off = 0;
    float*    hs   = (float*)(ws + off); off += (size_t)N_NODES * DIM * 4;  // 16MB
    float*    z    = (float*)(ws + off); off += (size_t)N_NODES * DIM * 4;  // 16MB
    char*     zb   = ws + off;                                              // zero region start
    float*    agg  = (float*)(ws + off); off += (size_t)N_NODES * DIM * 4;  // 16MB
    unsigned* mkey = (unsigned*)(ws + off); off += (size_t)N_NODES * 4;
    float*    den  = (float*)(ws + off); off += (size_t)N_NODES * 4;
    int*      deg  = (int*)(ws + off);   off += (size_t)N_NODES * 4;
    size_t    zbytes = (size_t)((ws + off) - zb);                           // zero region end
    float*    zsv  = (float*)(ws + off); off += (size_t)N_NODES * 4;
    float*    zdv  = (float*)(ws + off); off += (size_t)N_NODES * 4;
    float*    ev   = (float*)(ws + off); off += (size_t)N_EDGES * 4;
    float*    mv   = (float*)(ws + off); off += (size_t)N_NODES * 4;

    hipMemsetAsync(zb, 0, zbytes, stream);

    dim3 gGemm(N_NODES / 32, 2);   // 2 M-tiles/block, y: W_self vs W_func
    gat_gemm_dual<<<gGemm, 256, 0, stream>>>(h, Wself, Wfunc, hs, z);

    gat_node_scalars<<<N_NODES / 8, 256, 0, stream>>>(z, Watt, zsv, zdv);

    gat_edge_logits<<<N_EDGES / 8, 256, 0, stream>>>(edge_w, esrc, edst, Watt,
                                                     zsv, zdv, ev, mkey, deg);

    gat_decode_m<<<N_NODES / 256, 256, 0, stream>>>(mkey, deg, mv);

    gat_edge_scatter<<<N_EDGES / 8, 256, 0, stream>>>(ev, esrc, edst, mv, z,
                                                      den, agg);

    gat_finalize<<<N_NODES, 256, 0, stream>>>(h, hs, agg, den, deg, out);
}